// LanguageAttention_12489764897457
// MI455X (gfx1250) — compile-verified
//
#include <hip/hip_runtime.h>
#include <hip/hip_bf16.h>
#include <math.h>

// ---------------------------------------------------------------------------
// CDNA5 (gfx1250) fused multi-head attention, fp32 end-to-end via
// V_WMMA_F32_16X16X4_F32, with double-buffered Tensor-Data-Mover (TDM)
// LDS tile staging overlapped against WMMA on TENSORcnt.
// ---------------------------------------------------------------------------

typedef float v2f __attribute__((ext_vector_type(2)));
typedef float v8f __attribute__((ext_vector_type(8)));
typedef unsigned int u32x4 __attribute__((ext_vector_type(4)));
typedef int i32x8 __attribute__((ext_vector_type(8)));
typedef int i32x4 __attribute__((ext_vector_type(4)));

#define NEG_INF (-3.0e38f)

__device__ __forceinline__ v8f wmma_f32(v2f a, v2f b, v8f c) {
    // 8-arg pattern: (neg_a, A, neg_b, B, c_mod, C, reuse_a, reuse_b)
    return __builtin_amdgcn_wmma_f32_16x16x4_f32(false, a, false, b,
                                                 (short)0, c, false, false);
}

#if defined(__has_builtin)
#if __has_builtin(__builtin_amdgcn_tensor_load_to_lds)
#define HAVE_TDM 1
#endif
#endif

#ifdef HAVE_TDM
// Issue one TDM 2D tile load: tile_d1 rows of tile_d0 dwords from a row-major
// f32 tensor (row pitch stride0 dwords) into LDS at lds_addr, inserting
// pad dwords into LDS after every 2^(pad_ivl_code+1) dwords (ISA D# sec. 8).
__device__ __forceinline__ void tdm_load_tile_2d(
    unsigned lds_addr, const float* gaddr,
    unsigned tensor_d0, unsigned tile_d0, unsigned tile_d1,
    unsigned stride0, unsigned pad_ivl_code, unsigned pad_amt_code)
{
    unsigned long long ga = (unsigned long long)(uintptr_t)gaddr;
    u32x4 g0;
    g0.x = 1u;                                      // count=1 (valid), user mode
    g0.y = lds_addr;                                // LDS byte address
    g0.z = (unsigned)(ga & 0xFFFFFFFFu);            // global_addr [31:0]
    g0.w = (unsigned)((ga >> 32) & 0x01FFFFFFu)     // global_addr [56:32]
         | 0x80000000u;                             // type=2 ("image")
    i32x8 g1;
    g1[0] = (int)((2u << 16)                        // data_size = 4 bytes
                | (1u << 20)                        // pad_enable
                | (pad_ivl_code << 22)              // pad interval code
                | (pad_amt_code << 25));            // pad amount code
    g1[1] = (int)((tensor_d0 & 0xFFFFu) << 16);     // tensor_dim0[15:0]
    g1[2] = (int)((tensor_d0 >> 16)                 // tensor_dim0[31:16]
                | ((tile_d1 & 0xFFFFu) << 16));     // tensor_dim1[15:0]
    g1[3] = (int)((tile_d0 & 0xFFFFu) << 16);       // tensor_dim1[31:16]=0 | tile_dim0
    g1[4] = (int)(tile_d1 & 0xFFFFu);               // tile_dim1 (tile_dim2=0)
    g1[5] = (int)stride0;                           // tensor_dim0_stride[31:0]
    g1[6] = 0;                                      // stride0 hi | stride1 lo
    g1[7] = 0;                                      // stride1 hi
    i32x4 z4 = {0, 0, 0, 0};
    i32x8 z8 = {0, 0, 0, 0, 0, 0, 0, 0};
    // amdgpu-toolchain (clang-23) 6-arg form: (g0, g1, g2, g3, g4, cpol)
    __builtin_amdgcn_tensor_load_to_lds(g0, g1, z4, z4, z8, 0);
}
#endif

// ---------------------------------------------------------------------------
// GEMM: C[M,N] = A[M,K] @ W[K,N] + bias[N]     (M%128==0, N%128==0, K%32==0)
// 256 threads = 8 waves; block tile 128x128; wave tile 32x64 (2x4 WMMA accs).
// Double-buffered TDM staging: wave 0 keeps the next K-chunk's two tensor
// DMAs in flight (TENSORcnt<=2 wait) while all waves run WMMA on the
// current chunk.
// ---------------------------------------------------------------------------
__global__ __launch_bounds__(256)
void gemm_bias_kernel(const float* __restrict__ A, const float* __restrict__ W,
                      const float* __restrict__ bias, float* __restrict__ C,
                      int M, int N, int K) {
    __shared__ float Xs[2][128][36];   // 32-dword rows + 4 pad (TDM pad 4/32)
    __shared__ float Ws[2][32][132];   // 128-dword rows + 4 pad (TDM pad 4/128)

    const int tid  = threadIdx.x;
    const int wave = tid >> 5;
    const int lane = tid & 31;
    const int lm   = lane & 15;     // fragment row / col within 16
    const int lh   = lane >> 4;     // half-wave select (K pair)

    const int gm0 = blockIdx.x * 128;
    const int gn0 = blockIdx.y * 128;

    const int wm = wave & 3;        // 4 waves down M
    const int wn = wave >> 2;       // 2 waves across N
    const int m_base = wm * 32;
    const int n_base = wn * 64;

    const int nchunks = K >> 5;     // number of 32-wide K chunks

    v8f acc[2][4];
    #pragma unroll
    for (int mi = 0; mi < 2; ++mi)
        #pragma unroll
        for (int ni = 0; ni < 4; ++ni) acc[mi][ni] = (v8f){};

#ifdef HAVE_TDM
    // ---- prologue: start DMAs for chunks 0 and 1 ----
    if (wave == 0) {
        #pragma unroll
        for (int p = 0; p < 2; ++p) {
            if (p < nchunks) {
                int k0 = p * 32;
                tdm_load_tile_2d((unsigned)(uintptr_t)&Xs[p][0][0],
                                 &A[(size_t)gm0 * K + k0],
                                 (unsigned)K, 32u, 128u, (unsigned)K,
                                 4u /*32 dw*/, 3u /*pad 4 dw*/);
                tdm_load_tile_2d((unsigned)(uintptr_t)&Ws[p][0][0],
                                 &W[(size_t)k0 * N + gn0],
                                 (unsigned)N, 128u, 32u, (unsigned)N,
                                 6u /*128 dw*/, 3u /*pad 4 dw*/);
            }
        }
    }
#endif

    for (int ic = 0; ic < nchunks; ++ic) {
        const int buf = ic & 1;
#ifdef HAVE_TDM
        if (wave == 0) {
            // in-order TENSORcnt: <=2 outstanding means chunk ic has landed
            if (ic + 1 < nchunks) __builtin_amdgcn_s_wait_tensorcnt(2);
            else                  __builtin_amdgcn_s_wait_tensorcnt(0);
        }
#else
        {
            int k0 = ic * 32;
            #pragma unroll
            for (int it = 0; it < 4; ++it) {
                int t  = tid + it * 256;
                int m  = t >> 3;
                int k4 = (t & 7) * 4;
                float4 v = *(const float4*)&A[(size_t)(gm0 + m) * K + k0 + k4];
                *(float4*)&Xs[buf][m][k4] = v;
            }
            #pragma unroll
            for (int it = 0; it < 4; ++it) {
                int t  = tid + it * 256;
                int k  = t >> 5;
                int n4 = (t & 31) * 4;
                float4 v = *(const float4*)&W[(size_t)(k0 + k) * N + gn0 + n4];
                *(float4*)&Ws[buf][k][n4] = v;
            }
        }
#endif
        __syncthreads();

        #pragma unroll
        for (int kc = 0; kc < 32; kc += 4) {
            const int ka = kc + 2 * lh;         // K index of this lane's pair
            v2f afrag[2];
            #pragma unroll
            for (int mi = 0; mi < 2; ++mi)
                afrag[mi] = *(const v2f*)&Xs[buf][m_base + mi * 16 + lm][ka];
            v2f bfrag[4];
            #pragma unroll
            for (int ni = 0; ni < 4; ++ni) {
                int n = n_base + ni * 16 + lm;
                bfrag[ni].x = Ws[buf][ka + 0][n];
                bfrag[ni].y = Ws[buf][ka + 1][n];
            }
            #pragma unroll
            for (int mi = 0; mi < 2; ++mi)
                #pragma unroll
                for (int ni = 0; ni < 4; ++ni)
                    acc[mi][ni] = wmma_f32(afrag[mi], bfrag[ni], acc[mi][ni]);
        }
        __syncthreads();   // all waves done reading buf before it is refilled

#ifdef HAVE_TDM
        if (wave == 0 && ic + 2 < nchunks) {
            int k0 = (ic + 2) * 32;
            tdm_load_tile_2d((unsigned)(uintptr_t)&Xs[buf][0][0],
                             &A[(size_t)gm0 * K + k0],
                             (unsigned)K, 32u, 128u, (unsigned)K,
                             4u /*32 dw*/, 3u /*pad 4 dw*/);
            tdm_load_tile_2d((unsigned)(uintptr_t)&Ws[buf][0][0],
                             &W[(size_t)k0 * N + gn0],
                             (unsigned)N, 128u, 32u, (unsigned)N,
                             6u /*128 dw*/, 3u /*pad 4 dw*/);
        }
#endif
    }

    // ---- epilogue: bias + store ----
    #pragma unroll
    for (int mi = 0; mi < 2; ++mi) {
        #pragma unroll
        for (int ni = 0; ni < 4; ++ni) {
            int gn = gn0 + n_base + ni * 16 + lm;
            float bv = bias[gn];
            #pragma unroll
            for (int r = 0; r < 8; ++r) {
                int gm = gm0 + m_base + mi * 16 + r + 8 * lh;
                C[(size_t)gm * N + gn] = acc[mi][ni][r] + bv;
            }
        }
    }
}

// ---------------------------------------------------------------------------
// Fused attention: one block per (b, h, 16-row q tile).
//   qkv layout: [B*S, 1536] row-major; cols 0..511=Q, 512..1023=K, 1024..=V,
//   head h occupies 64 cols at h*64 inside each third.
// Phase 1: scores = Q K^T * scale  (+ causal & attn mask)  -> LDS
// Phase 2: row softmax in LDS, fused atomicAdd of P/8 into head-mean output
// Phase 3: O = P V via WMMA (4 waves x 16 cols)
// ---------------------------------------------------------------------------
#define S_LEN   2048
#define LDQKV   1536
#define PS_LD   2052            // 2048 + 4 pad (bank-safe strided reads)
#define QS_LD   68              // 64 + 4 pad

__global__ __launch_bounds__(256)
void attn_kernel(const float* __restrict__ qkv, const int* __restrict__ amask,
                 float* __restrict__ attn_out, float* __restrict__ attn_mean) {
    extern __shared__ float smem[];
    float* Ps  = smem;                  // 16 * PS_LD
    float* Qs  = Ps + 16 * PS_LD;       // 16 * QS_LD
    float* red = Qs + 16 * QS_LD;       // 16 * 17

    const int tid  = threadIdx.x;
    const int wave = tid >> 5;
    const int lane = tid & 31;
    const int lm   = lane & 15;
    const int lh   = lane >> 4;

    const int bx = blockIdx.x;
    const int b  = bx >> 10;            // / (8*128)
    const int h  = (bx >> 7) & 7;
    const int qt = bx & 127;
    const int q0 = qt * 16;

    const int nkt  = qt + 1;            // causal: only key tiles <= q tile
    const int klim = nkt * 16;

    // ---- load Q tile [16 x 64] into LDS ----
    {
        int m  = tid >> 4;
        int d4 = (tid & 15) * 4;
        float4 v = *(const float4*)&qkv[(size_t)(b * S_LEN + q0 + m) * LDQKV + h * 64 + d4];
        *(float4*)&Qs[m * QS_LD + d4] = v;
    }
    __syncthreads();

    // ---- Phase 1: scores ----
    for (int kt = wave; kt < nkt; kt += 8) {
        const int kn0 = kt * 16;
        const int key = kn0 + lm;       // this lane's key column
        v8f acc = (v8f){};
        const float* kbase = &qkv[(size_t)(b * S_LEN + key) * LDQKV + 512 + h * 64];
        #pragma unroll
        for (int kd = 0; kd < 64; kd += 4) {
            int ka = kd + 2 * lh;
            v2f a  = *(const v2f*)&Qs[lm * QS_LD + ka];
            v2f bb = *(const v2f*)&kbase[ka];
            acc = wmma_f32(a, bb, acc);
        }
        const bool mk = (amask[b * S_LEN + key] != 0);
        #pragma unroll
        for (int r = 0; r < 8; ++r) {
            int m = r + 8 * lh;
            int q = q0 + m;
            float val = (mk && key <= q) ? acc[r] * 0.125f : NEG_INF; // 64^-0.5
            Ps[m * PS_LD + key] = val;
        }
    }
    __syncthreads();

    // ---- Phase 2: softmax + fused mean accumulation ----
    {
        const int r   = tid >> 4;       // row 0..15
        const int sub = tid & 15;       // 16 threads per row
        float* prow = &Ps[r * PS_LD];

        float mx = NEG_INF;
        for (int k = sub; k < klim; k += 16) mx = fmaxf(mx, prow[k]);
        red[r * 17 + sub] = mx;
        __syncthreads();
        float rowmax = NEG_INF;
        #pragma unroll
        for (int i = 0; i < 16; ++i) rowmax = fmaxf(rowmax, red[r * 17 + i]);

        float s = 0.f;
        for (int k = sub; k < klim; k += 16) {
            float e = __expf(prow[k] - rowmax);
            prow[k] = e;
            s += e;
        }
        __syncthreads();                // red reads done before overwrite
        red[r * 17 + sub] = s;
        __syncthreads();
        float rowsum = 0.f;
        #pragma unroll
        for (int i = 0; i < 16; ++i) rowsum += red[r * 17 + i];
        float inv = 1.0f / rowsum;

        float* meanrow = &attn_mean[(size_t)(b * S_LEN + q0 + r) * S_LEN];
        for (int k = sub; k < klim; k += 16) {
            float p = prow[k] * inv;
            prow[k] = p;
            atomicAdd(&meanrow[k], 0.125f * p);   // mean over 8 heads
        }
    }
    __syncthreads();

    // ---- Phase 3: O = P @ V  (waves 0..3, one 16x16 output tile each) ----
    if (wave < 4) {
        const int n0 = wave * 16;
        v8f acc = (v8f){};
        const float* vbase = &qkv[(size_t)(b * S_LEN) * LDQKV + 1024 + h * 64 + n0 + lm];
        for (int kb = 0; kb < klim; kb += 4) {
            int ka = kb + 2 * lh;
            v2f a = *(const v2f*)&Ps[lm * PS_LD + ka];
            v2f bb;
            bb.x = vbase[(size_t)(ka + 0) * LDQKV];
            bb.y = vbase[(size_t)(ka + 1) * LDQKV];
            acc = wmma_f32(a, bb, acc);
        }
        #pragma unroll
        for (int r = 0; r < 8; ++r) {
            int m = r + 8 * lh;
            attn_out[(size_t)(b * S_LEN + q0 + m) * 512 + h * 64 + n0 + lm] = acc[r];
        }
    }
}

// ---------------------------------------------------------------------------
extern "C" void kernel_launch(void* const* d_in, const int* in_sizes, int n_in,
                              void* d_out, int out_size, void* d_ws, size_t ws_size,
                              hipStream_t stream) {
    (void)in_sizes; (void)n_in; (void)out_size; (void)ws_size;

    const float* x     = (const float*)d_in[0];   // [4,2048,512]
    const int*   amask = (const int*)  d_in[1];   // [4,2048]
    const float* Wqkv  = (const float*)d_in[2];   // [512,1536]
    const float* bqkv  = (const float*)d_in[3];   // [1536]
    const float* Wout  = (const float*)d_in[4];   // [512,512]
    const float* bout  = (const float*)d_in[5];   // [512]

    float* out  = (float*)d_out;                  // [4,2048,512]
    float* mean = out + (size_t)4 * 2048 * 512;   // [4,2048,2048]

    float* qkv_ws  = (float*)d_ws;                // [8192,1536]  (50.3 MB)
    float* attn_ws = qkv_ws + (size_t)8192 * 1536;// [8192,512]   (16.8 MB)

    // zero the head-mean accumulator (atomicAdd target)
    (void)hipMemsetAsync(mean, 0, (size_t)4 * 2048 * 2048 * sizeof(float), stream);

    // 1) QKV projection: [8192,512] @ [512,1536] + bqkv
    gemm_bias_kernel<<<dim3(8192 / 128, 1536 / 128), 256, 0, stream>>>(
        x, Wqkv, bqkv, qkv_ws, 8192, 1536, 512);

    // 2) fused attention (+ head-mean): one block per (b,h,q-tile)
    constexpr size_t SMEM = (size_t)(16 * PS_LD + 16 * QS_LD + 16 * 17) * sizeof(float);
    attn_kernel<<<4 * 8 * 128, 256, SMEM, stream>>>(qkv_ws, amask, attn_ws, mean);

    // 3) output projection: [8192,512] @ [512,512] + bout
    gemm_bias_kernel<<<dim3(8192 / 128, 512 / 128), 256, 0, stream>>>(
        attn_ws, Wout, bout, out, 8192, 512, 512);
}